// FNO_layer_64381559767120
// MI455X (gfx1250) — compile-verified
//
#include <hip/hip_runtime.h>
#include <math.h>

// ---------------------------------------------------------------------------
// FNO layer for MI455X (gfx1250), fp32 end-to-end via V_WMMA_F32_16X16X4_F32.
//
//  out = sin(0.2 * (x@W + B + irfft(lowpass_spectral_mix(rfft(x)[:64]))))
//
// Only 64 frequency bins survive, so rfft/irfft collapse to GEMMs against a
// 65536x128 cos/sin table T.  All big GEMMs run on f32 WMMA (exact fp32).
// This round: B-operands (W, Bpp) are K-pairwise packed so every WMMA B
// fragment is one aligned b64 load, and each wave owns 4 output tiles so one
// A fragment feeds 4 wmma (load:wmma ratio 5:4 instead of ~5:2).
// ---------------------------------------------------------------------------

typedef float v2f __attribute__((ext_vector_type(2)));
typedef float v8f __attribute__((ext_vector_type(8)));

#define N_PTS 65536
#define DDIM  256
#define KFREQ 64
#define NROW  128                  // 2*KFREQ rows (cos,sin interleaved)
#define NCHUNK 64                  // K-split of the 65536-deep DFT reduction
#define CHUNK_N (N_PTS / NCHUNK)   // 1024

// D = A(16x4) * B(4x16) + C, fp32.
// Fragment layout (ISA 7.12.2): A v2f = {A[lane%16][K0], A[lane%16][K0+1]},
// K0 = 0 (lanes 0-15) / 2 (lanes 16-31).  B assumed analogous: vgpr0 = rows
// K=0/2, vgpr1 = rows K=1/3.  C/D v8f: M = v (+8 for hi lanes), N = lane%16.
__device__ __forceinline__ v8f wmma_f32(v2f a, v2f b, v8f c) {
  return __builtin_amdgcn_wmma_f32_16x16x4_f32(false, a, false, b,
                                               (short)0, c, false, false);
}

// ---------------------------------------------------------------------------
// Kernel 1: trig table. T[n*128 + 2k] = cos(2*pi*k*n/N), +1 -> sin.
// Integer phase (k*n) mod 65536 is exact, then one accurate sincosf.
// ---------------------------------------------------------------------------
__global__ __launch_bounds__(256) void tgen_kernel(float* __restrict__ T) {
  int idx = blockIdx.x * 256 + threadIdx.x;     // 65536*64 total
  int n = idx >> 6;
  int k = idx & 63;
  unsigned m = ((unsigned)n * (unsigned)k) & 65535u;
  float ang = (float)m * 9.58737992429e-5f;     // 2*pi/65536
  float s, c;
  sincosf(ang, &s, &c);
  size_t base = (size_t)n * NROW + 2 * k;
  T[base]     = c;
  T[base + 1] = s;
}

// ---------------------------------------------------------------------------
// Kernel 2: K-pairwise pack of W:  Wp[(k/2)*256+col][2] = {W[k], W[k+1]}.
// Makes every WMMA B fragment a single b64 load in the fused kernel.
// ---------------------------------------------------------------------------
__global__ __launch_bounds__(256) void pack_w_kernel(
    const float* __restrict__ W, float* __restrict__ Wp) {
  int idx = blockIdx.x * 256 + threadIdx.x;     // 0..32767 pair-slots
  int kpair = idx >> 8;
  int col   = idx & 255;
  Wp[(size_t)idx * 2]     = W[(size_t)(2 * kpair) * DDIM + col];
  Wp[(size_t)idx * 2 + 1] = W[(size_t)(2 * kpair + 1) * DDIM + col];
}

// ---------------------------------------------------------------------------
// Kernel 3: truncated forward DFT as WMMA GEMM.
//   Ppart[chunk][row][d] = sum_{n in chunk} T[n,row] * x[n,d]
// grid = (8 row-tiles, 64 n-chunks); block = 256 (8 waves); wave -> 32 d cols.
// ---------------------------------------------------------------------------
__global__ __launch_bounds__(256) void dft_low_kernel(
    const float* __restrict__ x, const float* __restrict__ T,
    float* __restrict__ Ppart) {
  const int lane    = threadIdx.x & 31;
  const int wave    = threadIdx.x >> 5;         // 0..7
  const int lr      = lane & 15;
  const int halfSel = (lane >> 4) << 1;         // 0 or 2 (A/B K-half select)
  const int row0    = blockIdx.x * 16;          // frequency-row tile
  const int nbase   = blockIdx.y * CHUNK_N;     // n chunk
  const int d0      = wave * 32;

  v8f acc0 = {0.f,0.f,0.f,0.f,0.f,0.f,0.f,0.f};
  v8f acc1 = acc0;

  const int arow = row0 + lr;                   // A row held by this lane
  for (int nk = nbase; nk < nbase + CHUNK_N; nk += 4) {
    const int na = nk + halfSel;
    v2f a;                                      // A[row, K] = T[n, row]
    a.x = T[(size_t)na * NROW + arow];
    a.y = T[(size_t)(na + 1) * NROW + arow];
    v2f b0, b1;                                 // B[K, col] = x[n, d]
    b0.x = x[(size_t)na * DDIM + d0 + lr];
    b0.y = x[(size_t)(na + 1) * DDIM + d0 + lr];
    b1.x = x[(size_t)na * DDIM + d0 + 16 + lr];
    b1.y = x[(size_t)(na + 1) * DDIM + d0 + 16 + lr];
    acc0 = wmma_f32(a, b0, acc0);
    acc1 = wmma_f32(a, b1, acc1);
  }

  const int mbase = (lane >> 4) << 3;           // C/D: M = v (+8 for hi lanes)
  const int col   = d0 + lr;
  size_t base = (size_t)blockIdx.y * (NROW * DDIM);
#pragma unroll
  for (int v = 0; v < 8; ++v) {
    int r = row0 + mbase + v;
    Ppart[base + (size_t)r * DDIM + col]      = acc0[v];
    Ppart[base + (size_t)r * DDIM + col + 16] = acc1[v];
  }
}

// ---------------------------------------------------------------------------
// Kernel 4: deterministic fixed-order reduction over the 64 chunks.
// ---------------------------------------------------------------------------
__global__ __launch_bounds__(256) void reduce_chunks_kernel(
    const float* __restrict__ Ppart, float* __restrict__ C) {
  int idx = blockIdx.x * 256 + threadIdx.x;     // 0..32767
  float s = 0.f;
#pragma unroll 4
  for (int c = 0; c < NCHUNK; ++c)
    s += Ppart[(size_t)c * (NROW * DDIM) + idx];
  C[idx] = s;
}

// ---------------------------------------------------------------------------
// Kernel 5: per-frequency complex mix  out_low[k] = x_low[k] @ Wc[k] + Bc[k],
// written directly in K-pairwise-packed irfft-B form:
//   BppP[(k*256+f)*2 + 0] =  scale_k * Re(out_low[k,f])   (scale_0=1 else 2)
//   BppP[(k*256+f)*2 + 1] = -scale_k * Im(out_low[k,f])
// x_low from raw DFT rows: Re = C[2k]/N, Im = -C[2k+1]/N.
// ---------------------------------------------------------------------------
__global__ __launch_bounds__(256) void spectral_mix_kernel(
    const float* __restrict__ C, const float* __restrict__ P_low,
    const float* __restrict__ B_low, float* __restrict__ BppP) {
  __shared__ float sXr[DDIM];
  __shared__ float sXi[DDIM];
  const int k = blockIdx.x;
  const int f = threadIdx.x;
  const float invN = 1.0f / (float)N_PTS;
  sXr[f] =  C[(size_t)(2 * k) * DDIM + f] * invN;
  sXi[f] = -C[(size_t)(2 * k + 1) * DDIM + f] * invN;
  __syncthreads();

  float ar = B_low[((size_t)k * DDIM + f) * 2 + 0];
  float ai = B_low[((size_t)k * DDIM + f) * 2 + 1];
  const float* Pk = P_low + (size_t)k * DDIM * DDIM * 2;
#pragma unroll 4
  for (int d = 0; d < DDIM; ++d) {
    float pr = Pk[((size_t)d * DDIM + f) * 2 + 0];
    float pi = Pk[((size_t)d * DDIM + f) * 2 + 1];
    float xr = sXr[d], xi = sXi[d];
    ar = fmaf(xr, pr, fmaf(-xi, pi, ar));     // Re(x*p)
    ai = fmaf(xr, pi, fmaf( xi, pr, ai));     // Im(x*p)
  }
  float sc = (k == 0) ? 1.0f : 2.0f;
  BppP[((size_t)k * DDIM + f) * 2 + 0] =  sc * ar;
  BppP[((size_t)k * DDIM + f) * 2 + 1] = -sc * ai;
}

// ---------------------------------------------------------------------------
// Kernel 6 (dominant, 13 GFLOP): fused  out = sin(0.2*(x@W + Bias + T@Bpp)).
// Block = 256 threads = 8 waves: waves 0-3 -> n-tile 2b, waves 4-7 -> 2b+1.
// Each wave owns 64 d-columns = 4 WMMA tiles; per K-step: 1 A b64 + 4 B b64
// loads feed 4 wmma.  K = 256 (x@W) + 128 (irfft via T).
// ---------------------------------------------------------------------------
__global__ __launch_bounds__(256) void fused_out_kernel(
    const float* __restrict__ x, const float* __restrict__ Wp,
    const float* __restrict__ Bias, const float* __restrict__ T,
    const float* __restrict__ BppP, float* __restrict__ out) {
  const int lane    = threadIdx.x & 31;
  const int wave    = threadIdx.x >> 5;
  const int lr      = lane & 15;
  const int halfSel = (lane >> 4) << 1;         // 0 or 2
  const int n0      = (blockIdx.x * 2 + (wave >> 2)) * 16;
  const int d0      = (wave & 3) * 64;
  const int arow    = n0 + lr;
  const int col     = d0 + lr;

  v8f acc[4];
#pragma unroll
  for (int t = 0; t < 4; ++t) acc[t] = (v8f){0.f,0.f,0.f,0.f,0.f,0.f,0.f,0.f};

  // ---- x @ W (K = 256) ----
  for (int k0 = 0; k0 < DDIM; k0 += 4) {
    const int kb = k0 + halfSel;                        // even
    v2f a = *(const v2f*)(x + (size_t)arow * DDIM + kb);        // 8B aligned
    const float* wrow = Wp + ((size_t)(kb >> 1) * DDIM + col) * 2;
#pragma unroll
    for (int t = 0; t < 4; ++t) {
      v2f b = *(const v2f*)(wrow + t * 32);             // {W[kb], W[kb+1]}
      acc[t] = wmma_f32(a, b, acc[t]);
    }
  }
  // ---- T @ Bpp (K = 128, the irfft) ----
  for (int r0 = 0; r0 < NROW; r0 += 4) {
    const int rb = r0 + halfSel;                        // even
    v2f a = *(const v2f*)(T + (size_t)arow * NROW + rb);        // 8B aligned
    const float* brow = BppP + ((size_t)(rb >> 1) * DDIM + col) * 2;
#pragma unroll
    for (int t = 0; t < 4; ++t) {
      v2f b = *(const v2f*)(brow + t * 32);
      acc[t] = wmma_f32(a, b, acc[t]);
    }
  }

  const int mbase = (lane >> 4) << 3;                   // C/D: M = v (+8 hi)
#pragma unroll
  for (int t = 0; t < 4; ++t) {
    const int c = col + t * 16;
    const float bias = Bias[c];
#pragma unroll
    for (int v = 0; v < 8; ++v) {
      int n = n0 + mbase + v;
      out[(size_t)n * DDIM + c] = sinf(0.2f * (acc[t][v] + bias));
    }
  }
}

// ---------------------------------------------------------------------------
extern "C" void kernel_launch(void* const* d_in, const int* in_sizes, int n_in,
                              void* d_out, int out_size, void* d_ws,
                              size_t ws_size, hipStream_t stream) {
  (void)in_sizes; (void)n_in; (void)out_size; (void)ws_size;
  const float* x     = (const float*)d_in[0];   // (65536,256)
  const float* P_low = (const float*)d_in[1];   // (64,256,256,2)
  const float* B_low = (const float*)d_in[2];   // (64,256,2)
  const float* W     = (const float*)d_in[3];   // (256,256)
  const float* Bias  = (const float*)d_in[4];   // (256,)
  float* out = (float*)d_out;

  float* ws    = (float*)d_ws;
  float* T     = ws;                                     // 65536*128  (32 MB)
  float* Ppart = T + (size_t)N_PTS * NROW;               // 64*128*256 ( 8 MB)
  float* C     = Ppart + (size_t)NCHUNK * NROW * DDIM;   // 128*256
  float* BppP  = C + NROW * DDIM;                        // 128*256 packed
  float* Wp    = BppP + NROW * DDIM;                     // 256*256 packed

  tgen_kernel<<<dim3((N_PTS * KFREQ) / 256), dim3(256), 0, stream>>>(T);
  pack_w_kernel<<<dim3((DDIM * DDIM / 2) / 256), dim3(256), 0, stream>>>(W, Wp);
  dft_low_kernel<<<dim3(NROW / 16, NCHUNK), dim3(256), 0, stream>>>(x, T, Ppart);
  reduce_chunks_kernel<<<dim3((NROW * DDIM) / 256), dim3(256), 0, stream>>>(Ppart, C);
  spectral_mix_kernel<<<dim3(KFREQ), dim3(DDIM), 0, stream>>>(C, P_low, B_low, BppP);
  fused_out_kernel<<<dim3(N_PTS / 32), dim3(256), 0, stream>>>(x, Wp, Bias, T, BppP, out);
}